// GNNUS_BaseModel_65274912965235
// MI455X (gfx1250) — compile-verified
//
#include <hip/hip_runtime.h>
#include <math.h>

// ---------------------------------------------------------------------------
// GNNUS_BaseModel on MI455X (gfx1250, wave32).
//  - fp32 WMMA 16x16x4 for every dense GEMM (K multiples of 4: 48/20/128/40)
//  - weights staged in LDS, zero-padded, pair-interleaved -> branchless
//    ds_load_b64 B-fragments in the WMMA inner loop
//  - graph propagate = gather * norm + fp32 global atomics (L2-resident tables)
//  - gcn_norm precomputed once per graph, reused across branches
// ---------------------------------------------------------------------------

typedef float v2f __attribute__((ext_vector_type(2)));
typedef float v8f __attribute__((ext_vector_type(8)));

#define T_DIM   48
#define F_LT    128
#define C_DIM   7
#define HID     20

// ------------------------- graph normalization -----------------------------

__global__ void deg_kernel(const int* __restrict__ col, const float* __restrict__ ew,
                           float* __restrict__ deg, int E) {
  int e = blockIdx.x * blockDim.x + threadIdx.x;
  if (e < E) unsafeAtomicAdd(&deg[col[e]], ew[e]);
}

__global__ void dis_kernel(float* __restrict__ deg, int n) {
  int i = blockIdx.x * blockDim.x + threadIdx.x;
  if (i < n) {
    float d = deg[i];
    deg[i] = d > 0.0f ? rsqrtf(d) : 0.0f;   // in-place: deg -> D^{-1/2}
  }
}

__global__ void norm_kernel(const int* __restrict__ row, const int* __restrict__ col,
                            const float* __restrict__ ew, const float* __restrict__ dis,
                            float* __restrict__ nrm, int E) {
  int e = blockIdx.x * blockDim.x + threadIdx.x;
  if (e < E) nrm[e] = dis[row[e]] * ew[e] * dis[col[e]];
}

// --------------------------- WMMA dense linear -----------------------------
// One wave computes a 16-row tile of Y = X @ W (+bias) with
// V_WMMA_F32_16X16X4_F32. KDIM/NT compile-time; F runtime (<= NT*16).
//
// fp32 A 16x4 layout: lanes 0-15 hold rows with K {k0,k0+1} in the 2 VGPRs;
// lanes 16-31 same rows with K {k0+2,k0+3}. B 4x16 mirrors the K split.
// C/D: VGPR v holds row v (lanes 0-15) and row v+8 (lanes 16-31).
//
// Weights live in LDS, zero-padded to NTP=NT*16 columns, stored as K-pairs:
//   s[(k/2)*NTP*2 + c*2 + (k&1)] = (c < F) ? W[k*F + c] : 0
// so a B fragment for (k0, half, col) is one aligned ds_load_b64.

template <int KDIM, int NT, bool HAS_I, bool HAS_R>
__global__ void wmma_linear_kernel(const float* __restrict__ X, int n,
                                   const float* __restrict__ Wi,
                                   const float* __restrict__ Wr,
                                   const float* __restrict__ bias,
                                   float* __restrict__ Yi,
                                   float* __restrict__ Yr, int F) {
  constexpr int NTP = NT * 16;
  constexpr int WELEM = KDIM * NTP;
  __shared__ float sWi[HAS_I ? WELEM : 1];
  __shared__ float sWr[HAS_R ? WELEM : 1];

  // cooperative stage + zero-pad (all threads; barrier before row guard)
  for (int idx = threadIdx.x; idx < WELEM; idx += blockDim.x) {
    const int k = idx / NTP;
    const int c = idx - k * NTP;
    const int o = ((k >> 1) * NTP + c) * 2 + (k & 1);
    const bool cv = c < F;
    if (HAS_I) sWi[o] = cv ? Wi[k * F + c] : 0.0f;
    if (HAS_R) sWr[o] = cv ? Wr[k * F + c] : 0.0f;
  }
  __syncthreads();

  const int lane = threadIdx.x & 31;
  const int wave = threadIdx.x >> 5;
  const int wavesPerBlock = blockDim.x >> 5;
  const long rowBase = (long)(blockIdx.x * wavesPerBlock + wave) * 16;
  if (rowBase >= n) return;   // after the barrier: safe

  const int half = lane >> 4;     // 0: K {k0,k0+1}; 1: K {k0+2,k0+3}
  const int l16 = lane & 15;

  long arow = rowBase + l16;
  if (arow >= n) arow = n - 1;    // clamp loads, mask stores
  const float* xptr = X + arow * (long)KDIM;

  v8f accI[NT], accR[NT];
#pragma unroll
  for (int t = 0; t < NT; ++t) {
    accI[t] = (v8f){0.f, 0.f, 0.f, 0.f, 0.f, 0.f, 0.f, 0.f};
    accR[t] = (v8f){0.f, 0.f, 0.f, 0.f, 0.f, 0.f, 0.f, 0.f};
  }

  for (int k0 = 0; k0 < KDIM; k0 += 4) {
    // A fragment: two consecutive floats per lane (8B aligned: even offset)
    v2f a = *(const v2f*)(xptr + k0 + 2 * half);
    const int p = (k0 >> 1) + half;
#pragma unroll
    for (int t = 0; t < NT; ++t) {
      const int c = t * 16 + l16;
      if (HAS_I) {
        v2f b = *(const v2f*)(&sWi[(p * NTP + c) * 2]);
        accI[t] = __builtin_amdgcn_wmma_f32_16x16x4_f32(
            false, a, false, b, (short)0, accI[t], false, false);
      }
      if (HAS_R) {
        v2f b = *(const v2f*)(&sWr[(p * NTP + c) * 2]);
        accR[t] = __builtin_amdgcn_wmma_f32_16x16x4_f32(
            false, a, false, b, (short)0, accR[t], false, false);
      }
    }
  }

#pragma unroll
  for (int t = 0; t < NT; ++t) {
    const int colv = t * 16 + l16;
    if (colv >= F) continue;
    const float bval = HAS_R ? bias[colv] : 0.0f;
#pragma unroll
    for (int v = 0; v < 8; ++v) {
      const long r = rowBase + v + 8 * half;
      if (r < n) {
        if (HAS_I) Yi[r * (long)F + colv] = accI[t][v];
        if (HAS_R) Yr[r * (long)F + colv] = accR[t][v] + bval;
      }
    }
  }
}

// ------------------------------ propagate ----------------------------------
// Agg[col[e]][:] += norm[e] * P[row[e]][:].  Node tables are L2-resident
// (<= 8 MB << 192 MB L2); fp32 atomics go to L2 atomic units.

template <int F>
__global__ void propagate_kernel(const int* __restrict__ row, const int* __restrict__ col,
                                 const float* __restrict__ nrm,
                                 const float* __restrict__ P,
                                 float* __restrict__ Agg, int E) {
  int e = blockIdx.x * blockDim.x + threadIdx.x;
  if (e >= E) return;
  const int r = row[e];
  const int c = col[e];
  const float w = nrm[e];
  const float* src = P + (long)r * F;
  float* dst = Agg + (long)c * F;
#pragma unroll
  for (int f = 0; f < F; ++f) unsafeAtomicAdd(&dst[f], w * src[f]);
}

// --------------------------- pointwise kernels -----------------------------

__global__ void gelu_elu_kernel(const float* __restrict__ in, float* __restrict__ out,
                                long cnt) {
  long i = (long)blockIdx.x * blockDim.x + threadIdx.x;
  if (i < cnt) {
    float x = in[i];
    float g = 0.5f * x * (1.0f + erff(x * 0.70710678118654752f));  // exact GELU
    out[i] = g > 0.0f ? g : expm1f(g);                              // ELU(alpha=1)
  }
}

__global__ void relu_kernel(float* __restrict__ buf, long cnt) {
  long i = (long)blockIdx.x * blockDim.x + threadIdx.x;
  if (i < cnt) buf[i] = fmaxf(buf[i], 0.0f);
}

__global__ void softmax7_kernel(const float* __restrict__ in, float* __restrict__ out,
                                int n, int do_relu, int accumulate) {
  int i = blockIdx.x * blockDim.x + threadIdx.x;
  if (i >= n) return;
  float v[C_DIM];
  float mx = -3.0e38f;
#pragma unroll
  for (int j = 0; j < C_DIM; ++j) {
    float x = in[(long)i * C_DIM + j];
    if (do_relu) x = fmaxf(x, 0.0f);
    v[j] = x;
    mx = fmaxf(mx, x);
  }
  float s = 0.0f;
#pragma unroll
  for (int j = 0; j < C_DIM; ++j) { v[j] = expf(v[j] - mx); s += v[j]; }
  const float inv = 1.0f / s;
#pragma unroll
  for (int j = 0; j < C_DIM; ++j) {
    const float o = v[j] * inv;
    if (accumulate) out[(long)i * C_DIM + j] += o;
    else            out[(long)i * C_DIM + j] = o;
  }
}

// out = (2*ll + 2*lt) @ Wll + bll + gnn_sum @ Wout + bout
__global__ void final_kernel(const float* __restrict__ ll, const float* __restrict__ lt,
                             const float* __restrict__ gnn,
                             const float* __restrict__ Wll, const float* __restrict__ bll,
                             const float* __restrict__ Wout, const float* __restrict__ bout,
                             float* __restrict__ out, int n) {
  int i = blockIdx.x * blockDim.x + threadIdx.x;
  if (i >= n) return;
  float a[C_DIM], g[C_DIM];
#pragma unroll
  for (int j = 0; j < C_DIM; ++j) {
    a[j] = 2.0f * (ll[(long)i * C_DIM + j] + lt[(long)i * C_DIM + j]);
    g[j] = gnn[(long)i * C_DIM + j];
  }
#pragma unroll
  for (int j = 0; j < C_DIM; ++j) {
    float acc = bll[j] + bout[j];
#pragma unroll
    for (int k = 0; k < C_DIM; ++k)
      acc += a[k] * Wll[k * C_DIM + j] + g[k] * Wout[k * C_DIM + j];
    out[(long)i * C_DIM + j] = acc;
  }
}

// ------------------------------- host side ---------------------------------

extern "C" void kernel_launch(void* const* d_in, const int* in_sizes, int n_in,
                              void* d_out, int out_size, void* d_ws, size_t ws_size,
                              hipStream_t stream) {
  const int E = in_sizes[1];
  const int n = in_sizes[8] / T_DIM;   // B*N

  // ---- input map -----------------------------------------------------------
  // 0..13 in setup_inputs() insertion order; params leaves in sorted-key
  // pytree order: dense_ll{W,b}, dense_lt{W1,W2,b1,b2}, distance[{Wi,Wr,b}x2],
  // duration, loc, out{W,b}, temporal, week, weekend.
  const int*   A_ei    = (const int*)d_in[0];
  const float* A_ew    = (const float*)d_in[1];
  const int*   Aw_ei   = (const int*)d_in[2];
  const float* Aw_ew   = (const float*)d_in[3];
  const int*   Awe_ei  = (const int*)d_in[4];
  const float* Awe_ew  = (const float*)d_in[5];
  const int*   L_ei    = (const int*)d_in[6];
  const float* L_ew    = (const float*)d_in[7];
  const float* x_t     = (const float*)d_in[8];
  const float* x_w     = (const float*)d_in[9];
  const float* x_we    = (const float*)d_in[10];
  const float* x_d     = (const float*)d_in[11];
  const float* x_du    = (const float*)d_in[12];
  const float* x_lt    = (const float*)d_in[13];
  const float* Wll     = (const float*)d_in[14];  // dense_ll.W
  const float* bll     = (const float*)d_in[15];  // dense_ll.b
  const float* W1      = (const float*)d_in[16];  // dense_lt.W1
  const float* W2      = (const float*)d_in[17];  // dense_lt.W2
  const float* b1      = (const float*)d_in[18];
  const float* b2      = (const float*)d_in[19];
  const float* Wout    = (const float*)d_in[38];  // out.W
  const float* bout    = (const float*)d_in[39];  // out.b
  // branch param base indices
  const int P_DIST = 20, P_DUR = 26, P_LOC = 32, P_TEMP = 40, P_WEEK = 46, P_WKND = 52;

  // ---- workspace carve -----------------------------------------------------
  float* w = (float*)d_ws;
  float* dis0   = w; w += (size_t)4 * n;      // D^{-1/2} for 4 graphs
  float* norm0  = w; w += (size_t)4 * E;      // gcn norm for 4 graphs
  float* tmp0   = w; w += (size_t)n * HID;    // x @ Wi      [n,20]
  float* agg0   = w; w += (size_t)n * HID;    // scatter acc [n,20]
  float* hbuf   = w; w += (size_t)n * 40;     // h [n,20] / dense hidden [n,40]
  float* tmp1   = w; w += (size_t)n * C_DIM;  // h @ Wi2     [n,7]
  float* agg1   = w; w += (size_t)n * C_DIM;  // scatter acc [n,7]
  float* gnnacc = w; w += (size_t)n * C_DIM;  // sum of 5 branch softmaxes
  float* out_ll = w; w += (size_t)n * C_DIM;
  float* out_lt = w; w += (size_t)n * C_DIM;
  (void)ws_size; (void)n_in; (void)out_size;

  const int BLK = 256;
  const int gE  = (E + BLK - 1) / BLK;
  const int gN  = (n + BLK - 1) / BLK;
  const int rowsPerBlock = (BLK / 32) * 16;   // 8 waves x 16 rows
  const int gLin = (n + rowsPerBlock - 1) / rowsPerBlock;

  // ---- gcn_norm for the 4 graphs ------------------------------------------
  const int* eis[4]   = {A_ei, Aw_ei, Awe_ei, L_ei};
  const float* ews[4] = {A_ew, Aw_ew, Awe_ew, L_ew};
  for (int g = 0; g < 4; ++g) {
    float* dis = dis0 + (size_t)g * n;
    hipMemsetAsync(dis, 0, (size_t)n * sizeof(float), stream);
    deg_kernel<<<gE, BLK, 0, stream>>>(eis[g] + E, ews[g], dis, E);
    dis_kernel<<<gN, BLK, 0, stream>>>(dis, n);
    norm_kernel<<<gE, BLK, 0, stream>>>(eis[g], eis[g] + E, ews[g], dis,
                                        norm0 + (size_t)g * E, E);
  }

  hipMemsetAsync(gnnacc, 0, (size_t)n * C_DIM * sizeof(float), stream);

  // ---- one ARMA-ARMA branch -----------------------------------------------
  auto branch = [&](const float* x, const int* ei, const float* nrm,
                    int pbase, float* dst, int accumulate, bool isLoc) {
    const float* Wi1 = (const float*)d_in[pbase + 0];
    const float* Wr1 = (const float*)d_in[pbase + 1];
    const float* bb1 = (const float*)d_in[pbase + 2];
    const float* Wi2 = (const float*)d_in[pbase + 3];
    const float* Wr2 = (const float*)d_in[pbase + 4];
    const float* bb2 = (const float*)d_in[pbase + 5];

    // ARMA layer 1: tmp0 = x@Wi ; agg0 = x@Wr + b (scatter init)
    if (isLoc)
      wmma_linear_kernel<F_LT, 2, true, true><<<gLin, BLK, 0, stream>>>(
          x, n, Wi1, Wr1, bb1, tmp0, agg0, HID);
    else
      wmma_linear_kernel<T_DIM, 2, true, true><<<gLin, BLK, 0, stream>>>(
          x, n, Wi1, Wr1, bb1, tmp0, agg0, HID);
    propagate_kernel<HID><<<gE, BLK, 0, stream>>>(ei, ei + E, nrm, tmp0, agg0, E);
    const long cnt0 = (long)n * HID;
    gelu_elu_kernel<<<(int)((cnt0 + BLK - 1) / BLK), BLK, 0, stream>>>(agg0, hbuf, cnt0);

    // ARMA layer 2 on h: tmp1 = h@Wi2 ; agg1 = h@Wr2 + b2
    wmma_linear_kernel<HID, 1, true, true><<<gLin, BLK, 0, stream>>>(
        hbuf, n, Wi2, Wr2, bb2, tmp1, agg1, C_DIM);
    propagate_kernel<C_DIM><<<gE, BLK, 0, stream>>>(ei, ei + E, nrm, tmp1, agg1, E);
    softmax7_kernel<<<gN, BLK, 0, stream>>>(agg1, dst, n, /*relu=*/1, accumulate);
  };

  const float* nA  = norm0 + (size_t)0 * E;
  const float* nW  = norm0 + (size_t)1 * E;
  const float* nWe = norm0 + (size_t)2 * E;
  const float* nL  = norm0 + (size_t)3 * E;

  branch(x_t,  A_ei,   nA,  P_TEMP, gnnacc, 1, false);
  branch(x_w,  Aw_ei,  nW,  P_WEEK, gnnacc, 1, false);
  branch(x_we, Awe_ei, nWe, P_WKND, gnnacc, 1, false);
  branch(x_d,  A_ei,   nA,  P_DIST, gnnacc, 1, false);
  branch(x_du, A_ei,   nA,  P_DUR,  gnnacc, 1, false);
  branch(x_lt, L_ei,   nL,  P_LOC,  out_ll, 0, true);

  // ---- dense_lt MLP: softmax(relu(x@W1+b1)@W2 + b2) -----------------------
  wmma_linear_kernel<F_LT, 3, false, true><<<gLin, BLK, 0, stream>>>(
      x_lt, n, nullptr, W1, b1, nullptr, hbuf, 40);
  const long cntH = (long)n * 40;
  relu_kernel<<<(int)((cntH + BLK - 1) / BLK), BLK, 0, stream>>>(hbuf, cntH);
  wmma_linear_kernel<40, 1, false, true><<<gLin, BLK, 0, stream>>>(
      hbuf, n, nullptr, W2, b2, nullptr, agg1, C_DIM);
  softmax7_kernel<<<gN, BLK, 0, stream>>>(agg1, out_lt, n, /*relu=*/0, 0);

  // ---- final combine -------------------------------------------------------
  final_kernel<<<gN, BLK, 0, stream>>>(out_ll, out_lt, gnnacc,
                                       Wll, bll, Wout, bout, (float*)d_out, n);
}